// FNS_88304527606518
// MI455X (gfx1250) — compile-verified
//
#include <hip/hip_runtime.h>

typedef __attribute__((ext_vector_type(2))) float v2f;
typedef __attribute__((ext_vector_type(8))) float v8f;
typedef __attribute__((ext_vector_type(4))) unsigned int u32x4;
typedef __attribute__((ext_vector_type(8))) int i32x8;
typedef __attribute__((ext_vector_type(4))) int i32x4;

#define LD 272                    // padded leading dim (multiple of 16, >= 257)
#define PLANE (272*272)
#define TN (LD/16)                // 17 tiles per dim
#define KCH 32                    // K-chunk staged to LDS per TDM transfer

constexpr int N1r   = 255;
constexpr int CROPd = 257;
constexpr int BATCH = 8;
constexpr int CHID  = 8;
constexpr float PIf = 3.14159265358979323846f;

// ---------------------------------------------------------------------------
// TDM: DMA a tile_h x tile_w (f32 elements) tile with row stride `stride`
// from global memory into LDS at byte offset lds_off. D# layout per CDNA5
// ISA 8.3/8.4 (group0: count|lds|gaddr|type=2; group1: dsize|dims|tile|stride).
// ---------------------------------------------------------------------------
__device__ inline void tdm_load_2d(unsigned lds_off, const float* gptr,
                                   unsigned tile_w, unsigned tile_h,
                                   unsigned stride_elems) {
#if __has_builtin(__builtin_amdgcn_tensor_load_to_lds)
  unsigned long long ga = (unsigned long long)(__SIZE_TYPE__)gptr;
  u32x4 g0;
  g0.x = 1u;                                          // count=1, user descriptor
  g0.y = lds_off;                                     // lds_addr (bytes)
  g0.z = (unsigned)(ga & 0xffffffffull);              // global_addr[31:0]
  g0.w = (unsigned)((ga >> 32) & 0x01ffffffull) | (2u << 30); // addr[56:32]|type=2
  unsigned long long st = (unsigned long long)stride_elems;
  i32x8 g1;
  g1[0] = (int)(2u << 16);                            // wg_mask=0, data_size=2 (4B)
  g1[1] = (int)((tile_w & 0xffffu) << 16);            // tensor_dim0 lo16 (==tile_w)
  g1[2] = (int)((tile_w >> 16) | ((tile_h & 0xffffu) << 16)); // dim0 hi | dim1 lo
  g1[3] = (int)((tile_h >> 16) | (tile_w << 16));     // dim1 hi | tile_dim0
  g1[4] = (int)(tile_h & 0xffffu);                    // tile_dim1 | tile_dim2=0
  g1[5] = (int)(st & 0xffffffffull);                  // dim0_stride lo32
  g1[6] = (int)((st >> 32) & 0xffffull);              // dim0_stride hi16
  g1[7] = 0;
  i32x4 z4 = {0, 0, 0, 0};
#if __clang_major__ >= 23
  i32x8 z8 = {};
  __builtin_amdgcn_tensor_load_to_lds(g0, g1, z4, z4, z8, 0);
#else
  __builtin_amdgcn_tensor_load_to_lds(g0, g1, z4, z4, 0);
#endif
#else
  // Fallback: no-op, nothing is copied (not expected on this toolchain)
  (void)lds_off; (void)gptr; (void)tile_w; (void)tile_h; (void)stride_elems;
#endif
}

__device__ inline void tdm_wait() {
#if __has_builtin(__builtin_amdgcn_s_wait_tensorcnt)
  __builtin_amdgcn_s_wait_tensorcnt(0);
#endif
}

// ---------------------------------------------------------------------------
// Build DST matrix S[t,m] = sin(pi*(m+1)*(t-128)/256), t<257, m<255
// ---------------------------------------------------------------------------
__global__ void k_build_S(float* __restrict__ S) {
  int idx = blockIdx.x * blockDim.x + threadIdx.x;
  if (idx >= PLANE) return;
  int t = idx / LD, m = idx % LD;
  float v = 0.f;
  if (t < CROPd && m < N1r)
    v = sinf(PIf * (float)(m + 1) * (float)(t - 128) * (1.0f / 256.0f));
  S[idx] = v;
}

// ---------------------------------------------------------------------------
// Build DFT matrix W[k,t] = exp(-2*pi*i*k*(t-127)/513), k<255, t<257
// ---------------------------------------------------------------------------
__global__ void k_build_W(float* __restrict__ Wre, float* __restrict__ Wim) {
  int idx = blockIdx.x * blockDim.x + threadIdx.x;
  if (idx >= PLANE) return;
  int k = idx / LD, t = idx % LD;
  float cr = 0.f, ci = 0.f;
  if (k < N1r && t < CROPd) {
    float ang = -2.0f * PIf * (float)k * (float)(t - 127) * (1.0f / 513.0f);
    cr = cosf(ang);
    ci = sinf(ang);
  }
  Wre[idx] = cr;
  Wim[idx] = ci;
}

// ---------------------------------------------------------------------------
// Import r (B,1,255,255) contiguous -> padded 272x272 planes
// ---------------------------------------------------------------------------
__global__ void k_import_r(const float* __restrict__ r, float* __restrict__ Rpad) {
  long idx = (long)blockIdx.x * blockDim.x + threadIdx.x;
  long total = (long)BATCH * N1r * N1r;
  if (idx >= total) return;
  int j = (int)(idx % N1r);
  long q = idx / N1r;
  int i = (int)(q % N1r);
  int b = (int)(q / N1r);
  Rpad[(long)b * PLANE + (long)i * LD + j] = r[idx];
}

// ---------------------------------------------------------------------------
// Batched WMMA f32 GEMM on 272x272 padded planes:
//   C[b] = beta*C[b] + alpha * A[b] * op(B[b]),  op = id (TRANSB=0) or T (1)
// One wave per 16x16 C tile. Per K-chunk (KCH): the wave's A and B tiles are
// DMA'd into LDS by the Tensor Data Mover (tensor_load_to_lds), waited with
// s_wait_tensorcnt, then consumed by V_WMMA_F32_16X16X4_F32 from LDS.
// Fragment layouts per CDNA5 ISA 7.12.2:
//   A 16x4 f32 : lane<16 -> M=lane, K=0..1 ; lane>=16 -> K=2..3
//   C/D 16x16  : VGPR v -> row v + 8*(lane>=16), col = lane&15
// Dynamic LDS: 2 x 16 x KCH floats per wave (A chunk + B chunk) = 4KB/wave.
// ---------------------------------------------------------------------------
template <int TRANSB>
__global__ __launch_bounds__(256) void k_gemm_wmma(
    const float* __restrict__ A, long sA,
    const float* __restrict__ B, long sB,
    float* __restrict__ C, long sC,
    int K, float alpha, float beta)
{
  extern __shared__ float lbuf[];

  int b = blockIdx.y;
  A += (long)b * sA;
  B += (long)b * sB;
  C += (long)b * sC;

  int wave = (int)(threadIdx.x >> 5);
  int lane = (int)(threadIdx.x & 31);
  int tile = blockIdx.x * 8 + wave;
  if (tile >= TN * TN) return;          // wave-uniform exit (EXEC stays all-1s)
  int tm = tile / TN, tn = tile % TN;
  int m0 = tm * 16, n0 = tn * 16;
  int lh = lane >> 4;                   // selects K-pair (A/B frag), row-pair (C)
  int ml = lane & 15;

  const unsigned waveBytes = 2u * 16u * KCH * 4u;
  const unsigned ldsA_off = (unsigned)wave * waveBytes;
  const unsigned ldsB_off = ldsA_off + 16u * KCH * 4u;
  float* la = lbuf + (size_t)wave * (2 * 16 * KCH);
  float* lb = la + 16 * KCH;

  v8f acc = {};
  for (int kc = 0; kc < K; kc += KCH) {
    int kw = (K - kc < KCH) ? (K - kc) : KCH;   // 32 or 16 (K is 256 or 272)

    // DMA this chunk's A (16 rows x kw) and B tile into LDS via TDM.
    tdm_load_2d(ldsA_off, A + (long)m0 * LD + kc, (unsigned)kw, 16u, LD);
    if (TRANSB)
      tdm_load_2d(ldsB_off, B + (long)n0 * LD + kc, (unsigned)kw, 16u, LD);
    else
      tdm_load_2d(ldsB_off, B + (long)kc * LD + n0, 16u, (unsigned)kw, LD);
    tdm_wait();

    for (int k0 = 0; k0 < kw; k0 += 4) {
      int ka = k0 + 2 * lh;
      v2f af, bf;
      af.x = la[ml * kw + ka];
      af.y = la[ml * kw + ka + 1];
      if (TRANSB) {
        bf.x = lb[ml * kw + ka];
        bf.y = lb[ml * kw + ka + 1];
      } else {
        bf.x = lb[ka * 16 + ml];
        bf.y = lb[(ka + 1) * 16 + ml];
      }
      acc = __builtin_amdgcn_wmma_f32_16x16x4_f32(
          /*neg_a=*/false, af, /*neg_b=*/false, bf,
          /*c_mod=*/(short)0, acc, /*reuse_a=*/false, /*reuse_b=*/false);
    }
  }

#pragma unroll
  for (int v = 0; v < 8; ++v) {
    int row = m0 + v + 8 * lh;
    float* cp = C + (long)row * LD + n0 + ml;
    float prev = (beta != 0.0f) ? *cp : 0.0f;
    *cp = beta * prev + alpha * acc[v];
  }
}

// ---------------------------------------------------------------------------
// Per-sample complex 3x3 conv on 257x257 planes (272 stride).
// Weights staged in LDS. transposed=1 -> use conj(w[b,ci,co,dj,di]).
// ---------------------------------------------------------------------------
__global__ __launch_bounds__(256) void k_conv3x3(
    const float* __restrict__ xre, const float* __restrict__ xim, long sX,
    const float* __restrict__ wre, const float* __restrict__ wim,
    int cin, int cout, int transposed,
    float* __restrict__ yre, float* __restrict__ yim, long sY)
{
  __shared__ float swr[CHID * 9];
  __shared__ float swi[CHID * 9];
  int b  = blockIdx.z;
  int co = blockIdx.y;

  int t = threadIdx.x;
  if (t < cin * 9) {
    int ci = t / 9, r9 = t % 9;
    int di = r9 / 3, dj = r9 % 3;
    long widx;
    if (!transposed)
      widx = ((((long)b * cout + co) * cin + ci) * 3 + di) * 3 + dj;
    else
      widx = ((((long)b * cin + ci) * cout + co) * 3 + dj) * 3 + di;
    swr[t] = wre[widx];
    float wiv = wim[widx];
    swi[t] = transposed ? -wiv : wiv;
  }
  __syncthreads();

  int p = blockIdx.x * blockDim.x + threadIdx.x;
  if (p >= CROPd * CROPd) return;
  int i = p / CROPd, j = p % CROPd;

  float ar = 0.f, ai = 0.f;
  for (int ci = 0; ci < cin; ++ci) {
    const float* xr = xre + (long)b * sX + (long)ci * PLANE;
    const float* xi = xim + (long)b * sX + (long)ci * PLANE;
    __builtin_prefetch(xr + (long)i * LD + j, 0, 1);   // global_prefetch_b8
#pragma unroll
    for (int di = 0; di < 3; ++di) {
      int ii = i + di - 1;
      if (ii < 0 || ii >= CROPd) continue;
#pragma unroll
      for (int dj = 0; dj < 3; ++dj) {
        int jj = j + dj - 1;
        if (jj < 0 || jj >= CROPd) continue;
        float vr = xr[(long)ii * LD + jj];
        float vi = xi[(long)ii * LD + jj];
        float wr = swr[ci * 9 + di * 3 + dj];
        float wi = swi[ci * 9 + di * 3 + dj];
        ar = fmaf(vr, wr, fmaf(-vi, wi, ar));
        ai = fmaf(vr, wi, fmaf(vi, wr, ai));
      }
    }
  }
  long po = (long)b * sY + (long)co * PLANE + (long)i * LD + j;
  yre[po] = ar;
  yim[po] = ai;
}

// ---------------------------------------------------------------------------
// In-place complex multiply by theta (B, CHID, 257, 257) contiguous
// ---------------------------------------------------------------------------
__global__ void k_theta(float* __restrict__ xre, float* __restrict__ xim,
                        const float* __restrict__ tre, const float* __restrict__ tim)
{
  long idx = (long)blockIdx.x * blockDim.x + threadIdx.x;
  long total = (long)BATCH * CHID * CROPd * CROPd;
  if (idx >= total) return;
  int j = (int)(idx % CROPd);
  long q = idx / CROPd;
  int i = (int)(q % CROPd);
  q /= CROPd;
  int ch = (int)(q % CHID);
  int b  = (int)(q / CHID);
  long po = ((long)b * CHID + ch) * PLANE + (long)i * LD + j;
  float xr = xre[po], xi = xim[po];
  float trv = tre[idx], tiv = tim[idx];
  xre[po] = xr * trv - xi * tiv;
  xim[po] = xr * tiv + xi * trv;
}

// ---------------------------------------------------------------------------
// Emit complex64-interleaved output (B,1,255,255)
// ---------------------------------------------------------------------------
__global__ void k_emit(const float* __restrict__ Ere, const float* __restrict__ Eim,
                       float* __restrict__ out)
{
  long idx = (long)blockIdx.x * blockDim.x + threadIdx.x;
  long total = (long)BATCH * N1r * N1r;
  if (idx >= total) return;
  int j = (int)(idx % N1r);
  long q = idx / N1r;
  int i = (int)(q % N1r);
  int b = (int)(q / N1r);
  long po = (long)b * PLANE + (long)i * LD + j;
  out[2 * idx + 0] = Ere[po];
  out[2 * idx + 1] = Eim[po];
}

// ---------------------------------------------------------------------------
extern "C" void kernel_launch(void* const* d_in, const int* in_sizes, int n_in,
                              void* d_out, int out_size, void* d_ws, size_t ws_size,
                              hipStream_t stream) {
  (void)in_sizes; (void)n_in; (void)out_size;
  const float* r     = (const float*)d_in[0];
  const float* w1re  = (const float*)d_in[1];
  const float* w1im  = (const float*)d_in[2];
  const float* w2re  = (const float*)d_in[3];
  const float* w2im  = (const float*)d_in[4];
  const float* w3re  = (const float*)d_in[5];
  const float* w3im  = (const float*)d_in[6];
  const float* thre  = (const float*)d_in[7];
  const float* thim  = (const float*)d_in[8];

  float* ws = (float*)d_ws;
  long o = 0;
  float* S    = ws + o; o += PLANE;          // DST matrix (shared)
  float* Wre  = ws + o; o += PLANE;          // DFT matrix re (shared)
  float* Wim  = ws + o; o += PLANE;          // DFT matrix im (shared)
  float* Rpad = ws + o; o += (long)BATCH * PLANE;
  float* T1   = ws + o; o += (long)BATCH * PLANE;
  float* RHre = ws + o; o += (long)BATCH * PLANE;
  float* RHim = ws + o; o += (long)BATCH * PLANE;   // stays zero (rh is real)
  float* Xre  = ws + o; o += (long)BATCH * CHID * PLANE;
  float* Xim  = ws + o; o += (long)BATCH * CHID * PLANE;
  float* Yre  = ws + o; o += (long)BATCH * CHID * PLANE;
  float* Yim  = ws + o; o += (long)BATCH * CHID * PLANE;
  float* Ure  = ws + o; o += (long)BATCH * PLANE;
  float* Uim  = ws + o; o += (long)BATCH * PLANE;
  float* Ere  = ws + o; o += (long)BATCH * PLANE;
  float* Eim  = ws + o; o += (long)BATCH * PLANE;
  size_t needed = (size_t)o * sizeof(float);
  if (ws_size < needed) return;

  hipMemsetAsync(d_ws, 0, needed, stream);

  const int B256 = 256;
  const size_t gemmLds = 8u * 2u * 16u * KCH * sizeof(float);  // 32 KB / block
  dim3 gplane((PLANE + B256 - 1) / B256);
  k_build_S<<<gplane, B256, 0, stream>>>(S);
  k_build_W<<<gplane, B256, 0, stream>>>(Wre, Wim);

  long nr = (long)BATCH * N1r * N1r;
  k_import_r<<<dim3((unsigned)((nr + B256 - 1) / B256)), B256, 0, stream>>>(r, Rpad);

  dim3 gg((TN * TN + 7) / 8, BATCH);   // 8 waves (tiles) per 256-thread block
  const float dstScale = -4.0f / (512.0f * 512.0f);

  // Stage A: rh = dstScale * S * (R * S^T)
  k_gemm_wmma<1><<<gg, B256, gemmLds, stream>>>(Rpad, PLANE, S, 0, T1, PLANE, 256, 1.0f, 0.0f);
  k_gemm_wmma<0><<<gg, B256, gemmLds, stream>>>(S, 0, T1, PLANE, RHre, PLANE, 256, dstScale, 0.0f);

  // Stage B: conv chain (RH -> X -> Y -> X), theta, transposed chain
  dim3 gc((CROPd * CROPd + B256 - 1) / B256, CHID, BATCH);
  long sT = (long)CHID * PLANE;
  k_conv3x3<<<gc, B256, 0, stream>>>(RHre, RHim, PLANE, w1re, w1im, 1, CHID, 0, Xre, Xim, sT);
  k_conv3x3<<<gc, B256, 0, stream>>>(Xre, Xim, sT, w2re, w2im, CHID, CHID, 0, Yre, Yim, sT);
  k_conv3x3<<<gc, B256, 0, stream>>>(Yre, Yim, sT, w3re, w3im, CHID, CHID, 0, Xre, Xim, sT);

  long nt = (long)BATCH * CHID * CROPd * CROPd;
  k_theta<<<dim3((unsigned)((nt + B256 - 1) / B256)), B256, 0, stream>>>(Xre, Xim, thre, thim);

  k_conv3x3<<<gc, B256, 0, stream>>>(Xre, Xim, sT, w3re, w3im, CHID, CHID, 1, Yre, Yim, sT);
  k_conv3x3<<<gc, B256, 0, stream>>>(Yre, Yim, sT, w2re, w2im, CHID, CHID, 1, Xre, Xim, sT);
  dim3 gc1((CROPd * CROPd + B256 - 1) / B256, 1, BATCH);
  k_conv3x3<<<gc1, B256, 0, stream>>>(Xre, Xim, sT, w1re, w1im, CHID, 1, 1, Yre, Yim, sT);

  // Stage C: E = W * (Z * W^T), complex via 8 real GEMMs. Z = Y channel 0.
  // U = Z * W^T
  k_gemm_wmma<1><<<gg, B256, gemmLds, stream>>>(Yre, sT, Wre, 0, Ure, PLANE, LD,  1.0f, 0.0f);
  k_gemm_wmma<1><<<gg, B256, gemmLds, stream>>>(Yim, sT, Wim, 0, Ure, PLANE, LD, -1.0f, 1.0f);
  k_gemm_wmma<1><<<gg, B256, gemmLds, stream>>>(Yre, sT, Wim, 0, Uim, PLANE, LD,  1.0f, 0.0f);
  k_gemm_wmma<1><<<gg, B256, gemmLds, stream>>>(Yim, sT, Wre, 0, Uim, PLANE, LD,  1.0f, 1.0f);
  // E = W * U
  k_gemm_wmma<0><<<gg, B256, gemmLds, stream>>>(Wre, 0, Ure, PLANE, Ere, PLANE, LD,  1.0f, 0.0f);
  k_gemm_wmma<0><<<gg, B256, gemmLds, stream>>>(Wim, 0, Uim, PLANE, Ere, PLANE, LD, -1.0f, 1.0f);
  k_gemm_wmma<0><<<gg, B256, gemmLds, stream>>>(Wre, 0, Uim, PLANE, Eim, PLANE, LD,  1.0f, 0.0f);
  k_gemm_wmma<0><<<gg, B256, gemmLds, stream>>>(Wim, 0, Ure, PLANE, Eim, PLANE, LD,  1.0f, 1.0f);

  k_emit<<<dim3((unsigned)((nr + B256 - 1) / B256)), B256, 0, stream>>>(Ere, Eim, (float*)d_out);
}